// ReservoirEmbedding_74251394613895
// MI455X (gfx1250) — compile-verified
//
#include <hip/hip_runtime.h>

// ReservoirEmbedding for MI455X (gfx1250, wave32).
// out[t, :] = sum_{r<8} mask(idx) * W[reservoir_lookup[base_indices[t], r], :]
// Pure gather+reduce, AI ~0.25 FLOP/B -> memory bound; no matrix op applies.
// Strategy: 1 token per wave32. Indices are wave-uniform -> scalar (SMEM)
// loads (s_load_b32 + s_load_b256) + SGPR address math; 8x global_load_b128
// in saddr form (uniform row base in SGPRs, lane*16B as 32-bit voffset);
// branchless frozen-row mask via scalar 0/1 FMA; non-temporal b128 store.

#define FEAT 128
#define RES 8
#define FROZEN 0
#define WAVES_PER_BLOCK 8   // 256 threads = 8 wave32

typedef __attribute__((ext_vector_type(4))) float v4f;
typedef __attribute__((ext_vector_type(4))) int   v4i;

__global__ __launch_bounds__(WAVES_PER_BLOCK * 32)
void reservoir_embed_kernel(const int* __restrict__ base_indices,
                            const int* __restrict__ reservoir_lookup,
                            const float* __restrict__ W,
                            float* __restrict__ out,
                            int n_tokens)
{
    const int lane = threadIdx.x & 31;
    // Force wave-uniformity so the whole index chain lives in SGPRs.
    const int wave = __builtin_amdgcn_readfirstlane((int)(threadIdx.x >> 5));
    const int t    = (int)blockIdx.x * WAVES_PER_BLOCK + wave;
    if (t >= n_tokens) return;

    // Uniform address -> s_load_b32 of the base vocab index.
    int base = base_indices[t];
    base = __builtin_amdgcn_readfirstlane(base);

    // 8 reservoir indices: 32B-aligned, uniform address -> s_load_b256.
    const int* lk = reservoir_lookup + (size_t)base * RES;
    const v4i ilo = *(const v4i*)(lk);
    const v4i ihi = *(const v4i*)(lk + 4);
    int idx[RES];
    idx[0] = __builtin_amdgcn_readfirstlane(ilo.x);
    idx[1] = __builtin_amdgcn_readfirstlane(ilo.y);
    idx[2] = __builtin_amdgcn_readfirstlane(ilo.z);
    idx[3] = __builtin_amdgcn_readfirstlane(ilo.w);
    idx[4] = __builtin_amdgcn_readfirstlane(ihi.x);
    idx[5] = __builtin_amdgcn_readfirstlane(ihi.y);
    idx[6] = __builtin_amdgcn_readfirstlane(ihi.z);
    idx[7] = __builtin_amdgcn_readfirstlane(ihi.w);

    // Issue all 8 gathers first (8-deep b128 batch), then reduce.
    // Association: UNIFORM row pointer + divergent lane index, so ISel picks
    // the global saddr form (SGPR base, 32-bit voffset) instead of building
    // 64-bit VGPR addresses per row.
    v4f rows[RES];
#pragma unroll
    for (int r = 0; r < RES; ++r) {
        const v4f* rowp = (const v4f*)(W + (size_t)idx[r] * FEAT);  // uniform
        rows[r] = rowp[lane];                                       // +lane*16B
    }

    v4f acc = (v4f)0.0f;
#pragma unroll
    for (int r = 0; r < RES; ++r) {
        // Scalar 0/1 mask (SGPR) -> 4 v_fma_f32 per row per lane.
        const float m = (idx[r] != FROZEN) ? 1.0f : 0.0f;
        acc += rows[r] * m;
    }

    // Write-once 16.8 MB output: non-temporal so it doesn't evict the
    // embedding table (25.7 MB, resident in the 192 MB L2).
    v4f* outp = (v4f*)(out + (size_t)t * FEAT);                     // uniform
    __builtin_nontemporal_store(acc, outp + lane);                  // +lane*16B
}

extern "C" void kernel_launch(void* const* d_in, const int* in_sizes, int n_in,
                              void* d_out, int out_size, void* d_ws, size_t ws_size,
                              hipStream_t stream) {
    const int*   base_indices     = (const int*)d_in[0];
    const int*   reservoir_lookup = (const int*)d_in[1];
    const float* embedding_weight = (const float*)d_in[2];
    float*       out              = (float*)d_out;

    const int n_tokens = out_size / FEAT;                      // 32768
    const int blocks   = (n_tokens + WAVES_PER_BLOCK - 1) / WAVES_PER_BLOCK;

    reservoir_embed_kernel<<<blocks, WAVES_PER_BLOCK * 32, 0, stream>>>(
        base_indices, reservoir_lookup, embedding_weight, out, n_tokens);
}